// DPCLoss_43490838839625
// MI455X (gfx1250) — compile-verified
//
#include <hip/hip_runtime.h>
#include <math.h>

typedef float v2f __attribute__((ext_vector_type(2)));
typedef float v8f __attribute__((ext_vector_type(8)));

#define CDIM 256
#define KSTEPS (CDIM / 4)          // 64 fp32-WMMA K steps
#define MROWS 8192
#define JT_TOTAL (MROWS / 16)      // 512 column tiles of 16
#define ROW_BLOCKS (MROWS / 16)    // 512 row tiles of 16
#define WAVES_PER_BLOCK 8

// Fused  P @ G^T-style GEMM + online logsumexp + diagonal extraction.
// pred/gt raw layout: elem(m, c) at (m>>4)*4096 + c*16 + (m&15)
__global__ __launch_bounds__(256) void dpc_loss_main(const float* __restrict__ pred,
                                                     const float* __restrict__ gt,
                                                     float* __restrict__ partial)
{
    __shared__ float lds_ms[WAVES_PER_BLOCK][16][2]; // per-wave (max, sumexp) per row
    __shared__ float lds_diag[16];
    __shared__ float lds_row[16];

    const int rb   = blockIdx.x;       // 16-row tile index
    const int tid  = threadIdx.x;
    const int wave = tid >> 5;
    const int lane = tid & 31;
    const int half = lane >> 4;        // wave32 halves
    const int lm   = lane & 15;

    // A fragment base: lane l, vgpr v -> pred[row = rb*16 + lm][k = 4*kk + 2*half + v]
    const float* aBase = pred + (size_t)rb * (CDIM * 16) + lm + half * 32;

    // Hoist the whole A row-block into registers: invariant across all j-tiles.
    v2f aReg[KSTEPS];
#pragma unroll
    for (int kk = 0; kk < KSTEPS; ++kk) {
        aReg[kk].x = aBase[kk * 64];        // K = 4*kk + 2*half
        aReg[kk].y = aBase[kk * 64 + 16];   // K = 4*kk + 2*half + 1
    }

    float rmax[8], rsum[8];
#pragma unroll
    for (int v = 0; v < 8; ++v) { rmax[v] = -INFINITY; rsum[v] = 0.0f; }

    for (int jt = wave; jt < JT_TOTAL; jt += WAVES_PER_BLOCK) {
        const float* bBase = gt + (size_t)jt * (CDIM * 16) + lm + half * 32;
        // Two interleaved accumulation chains for WMMA ILP.
        v8f acc0 = {}, acc1 = {};
#pragma unroll
        for (int kk = 0; kk < KSTEPS; kk += 2) {
            v2f b0, b1;
            b0.x = bBase[kk * 64];
            b0.y = bBase[kk * 64 + 16];
            b1.x = bBase[kk * 64 + 64];
            b1.y = bBase[kk * 64 + 80];
            acc0 = __builtin_amdgcn_wmma_f32_16x16x4_f32(false, aReg[kk], false, b0,
                                                         (short)0, acc0, false, false);
            acc1 = __builtin_amdgcn_wmma_f32_16x16x4_f32(false, aReg[kk + 1], false, b1,
                                                         (short)0, acc1, false, false);
        }
        v8f acc;
#pragma unroll
        for (int v = 0; v < 8; ++v) acc[v] = acc0[v] + acc1[v];

        // Diagonal lives in the j-tile matching this row tile.
        // acc[v] = logits[row = v + 8*half][col = jt*16 + lm]
        if (jt == rb) {
            int v = lm - half * 8;
            if (v >= 0 && v < 8) lds_diag[lm] = acc[v];
        }

        // Online logsumexp update (per lane: 8 rows, this lane's column subset)
#pragma unroll
        for (int v = 0; v < 8; ++v) {
            float x  = acc[v];
            float nm = fmaxf(rmax[v], x);
            rsum[v]  = rsum[v] * __expf(rmax[v] - nm) + __expf(x - nm);
            rmax[v]  = nm;
        }
    }

    // Reduce across the 16 lanes of each half (same rows, different columns).
    // xor masks 1,2,4,8 never cross the 16-lane halves.
#pragma unroll
    for (int v = 0; v < 8; ++v) {
        float m = rmax[v], s = rsum[v];
#pragma unroll
        for (int off = 1; off < 16; off <<= 1) {
            float m2 = __shfl_xor(m, off, 32);
            float s2 = __shfl_xor(s, off, 32);
            float nm = fmaxf(m, m2);
            s = s * __expf(m - nm) + s2 * __expf(m2 - nm);
            m = nm;
        }
        if (lm == 0) { // lane 0 -> rows 0..7, lane 16 -> rows 8..15
            lds_ms[wave][v + half * 8][0] = m;
            lds_ms[wave][v + half * 8][1] = s;
        }
    }
    __syncthreads();

    // Merge the 8 waves' partial (max,sum) per row; then lse - diag.
    if (tid < 16) {
        float m = lds_ms[0][tid][0], s = lds_ms[0][tid][1];
#pragma unroll
        for (int w = 1; w < WAVES_PER_BLOCK; ++w) {
            float m2 = lds_ms[w][tid][0], s2 = lds_ms[w][tid][1];
            float nm = fmaxf(m, m2);
            s = s * __expf(m - nm) + s2 * __expf(m2 - nm);
            m = nm;
        }
        lds_row[tid] = (m + __logf(s)) - lds_diag[tid];
    }
    __syncthreads();

    if (tid == 0) {
        float acc = 0.0f;
#pragma unroll
        for (int r = 0; r < 16; ++r) acc += lds_row[r];
        partial[rb] = acc;
    }
}

// Deterministic final reduction: 512 partials -> mean loss
__global__ __launch_bounds__(512) void dpc_loss_reduce(const float* __restrict__ partial,
                                                       float* __restrict__ out)
{
    __shared__ float sm[512];
    int t = threadIdx.x;
    sm[t] = partial[t];
    __syncthreads();
    for (int s = 256; s > 0; s >>= 1) {
        if (t < s) sm[t] += sm[t + s];
        __syncthreads();
    }
    if (t == 0) out[0] = sm[0] * (1.0f / (float)MROWS);
}

extern "C" void kernel_launch(void* const* d_in, const int* in_sizes, int n_in,
                              void* d_out, int out_size, void* d_ws, size_t ws_size,
                              hipStream_t stream) {
    const float* pred = (const float*)d_in[0];
    const float* gt   = (const float*)d_in[1];
    float* out        = (float*)d_out;
    float* partial    = (float*)d_ws;   // 512 floats

    dpc_loss_main<<<ROW_BLOCKS, 256, 0, stream>>>(pred, gt, partial);
    dpc_loss_reduce<<<1, 512, 0, stream>>>(partial, out);
}